// SampleAndGroupModule_88244398064374
// MI455X (gfx1250) — compile-verified
//
#include <hip/hip_runtime.h>
#include <hip/hip_bf16.h>

// Problem constants (match reference)
#define N_PTS  32768
#define S_SAMP 2048
#define K_NB   32
#define E_F    128
#define M_ROWS (S_SAMP * K_NB)   // 65536 rows through the MLP
#define EPS_BN 1e-5f

typedef __attribute__((ext_vector_type(2))) float v2f;
typedef __attribute__((ext_vector_type(8))) float v8f;

// Wave32 argmax/argmin reductions (no barriers, pure VALU + lane crossing).
__device__ __forceinline__ void wave_argmax(float& v, int& i) {
#pragma unroll
  for (int m = 16; m > 0; m >>= 1) {
    float ov = __shfl_xor(v, m, 32);
    int   oi = __shfl_xor(i, m, 32);
    if (ov > v || (ov == v && oi < i)) { v = ov; i = oi; }
  }
}
__device__ __forceinline__ void wave_argmin(float& v, int& i) {
#pragma unroll
  for (int m = 16; m > 0; m >>= 1) {
    float ov = __shfl_xor(v, m, 32);
    int   oi = __shfl_xor(i, m, 32);
    if (ov < v || (ov == v && oi < i)) { v = ov; i = oi; }
  }
}

// ---------------------------------------------------------------------------
// 0) Pack xyz = points[:, :3] contiguously (384 KB -> lives in L2)
// ---------------------------------------------------------------------------
__global__ void pack_xyz_kernel(const float* __restrict__ points,
                                float* __restrict__ xyz) {
  int i = blockIdx.x * blockDim.x + threadIdx.x;
  if (i < N_PTS) {
    xyz[i * 3 + 0] = points[i * 6 + 0];
    xyz[i * 3 + 1] = points[i * 6 + 1];
    xyz[i * 3 + 2] = points[i * 6 + 2];
  }
}

// ---------------------------------------------------------------------------
// 1) Farthest point sampling: single 1024-thread workgroup, step loop inside.
//    Each thread owns a 32-element slice of d_all held entirely in VGPRs
//    (fully unrolled -> constant register indices, no scratch, no global
//    round trips). Per-step argmax: wave shfl reduction + one 32-entry LDS
//    pass -> only 2 barriers per step.
// ---------------------------------------------------------------------------
__global__ void __launch_bounds__(1024)
fps_kernel(const float* __restrict__ xyz, int* __restrict__ ids,
           const int* __restrict__ first_idx) {
  __shared__ float s_wv[32];
  __shared__ int   s_wi[32];
  __shared__ int   s_cur;
  const int tid  = threadIdx.x;
  const int lane = tid & 31;
  const int wid  = tid >> 5;

  float dreg[32];
#pragma unroll
  for (int k = 0; k < 32; ++k) dreg[k] = __builtin_inff();

  int cur = first_idx[0];
  if (tid == 0) ids[0] = cur;

  for (int step = 1; step < S_SAMP; ++step) {
    // Broadcast load of the current centroid (same address on all lanes).
    const float cx = xyz[cur * 3 + 0];
    const float cy = xyz[cur * 3 + 1];
    const float cz = xyz[cur * 3 + 2];

    float best = -__builtin_inff();
    int   bi   = 0;
#pragma unroll
    for (int k = 0; k < 32; ++k) {
      const int i = tid + (k << 10);          // strided -> coalesced loads
      float dx = xyz[i * 3 + 0] - cx;
      float dy = xyz[i * 3 + 1] - cy;
      float dz = xyz[i * 3 + 2] - cz;
      float d  = dx * dx + dy * dy + dz * dz;
      float da = fminf(dreg[k], d);
      if (i == cur) da = -__builtin_inff();
      dreg[k] = da;
      if (da > best) { best = da; bi = i; }   // strict > keeps lowest index
    }

    wave_argmax(best, bi);
    if (lane == 0) { s_wv[wid] = best; s_wi[wid] = bi; }
    __syncthreads();
    if (wid == 0) {
      float v = s_wv[lane];
      int   i = s_wi[lane];
      wave_argmax(v, i);                      // jnp.argmax tie-break: low idx
      if (lane == 0) { s_cur = i; ids[step] = i; }
    }
    __syncthreads();
    cur = s_cur;
  }
}

// ---------------------------------------------------------------------------
// 2) Per-centroid distances + exact top-K (stand-in for approx_max_k).
//    Full 32768-float distance row in LDS (128 KB of the 320 KB/WGP);
//    K argmin passes with wave-level reduction (2 barriers per pass).
// ---------------------------------------------------------------------------
__global__ void __launch_bounds__(256)
topk_kernel(const float* __restrict__ xyz, const int* __restrict__ ids,
            int* __restrict__ groups) {
  extern __shared__ float s_dist[];      // N_PTS floats (dynamic, 128 KB)
  __shared__ float s_wv[8];
  __shared__ int   s_wi[8];
  const int s    = blockIdx.x;
  const int tid  = threadIdx.x;
  const int lane = tid & 31;
  const int wid  = tid >> 5;
  const int c    = ids[s];
  const float cx = xyz[c * 3 + 0], cy = xyz[c * 3 + 1], cz = xyz[c * 3 + 2];

  for (int i = tid; i < N_PTS; i += 256) {
    float dx = xyz[i * 3 + 0] - cx;
    float dy = xyz[i * 3 + 1] - cy;
    float dz = xyz[i * 3 + 2] - cz;
    s_dist[i] = dx * dx + dy * dy + dz * dz;
  }
  __syncthreads();

  for (int k = 0; k < K_NB; ++k) {
    float best = __builtin_inff();
    int   bi   = 0;
    for (int i = tid; i < N_PTS; i += 256) {
      float d = s_dist[i];
      if (d < best) { best = d; bi = i; }
    }
    wave_argmin(best, bi);
    if (lane == 0) { s_wv[wid] = best; s_wi[wid] = bi; }
    __syncthreads();
    if (tid == 0) {
      float v = s_wv[0];
      int   w = s_wi[0];
#pragma unroll
      for (int j = 1; j < 8; ++j) {
        if (s_wv[j] < v || (s_wv[j] == v && s_wi[j] < w)) {
          v = s_wv[j]; w = s_wi[j];
        }
      }
      groups[s * K_NB + k] = w;
      s_dist[w] = __builtin_inff();
    }
    __syncthreads();
  }
}

// ---------------------------------------------------------------------------
// 3) Gather features: feats[s,k,0:6] = points[g] - float(ids[s]);
//                     feats[s,k,6:12] = points[g]
// ---------------------------------------------------------------------------
__global__ void feats_kernel(const float* __restrict__ points,
                             const int* __restrict__ ids,
                             const int* __restrict__ groups,
                             float* __restrict__ feats) {
  int t = blockIdx.x * blockDim.x + threadIdx.x;
  if (t >= M_ROWS) return;
  const int   s   = t / K_NB;
  const int   g   = groups[t];
  const float idv = (float)ids[s];
  const float* p  = points + (size_t)g * 6;
  float* f = feats + (size_t)t * 12;
#pragma unroll
  for (int j = 0; j < 6; ++j) {
    float pv = p[j];
    f[j]     = pv - idv;
    f[6 + j] = pv;
  }
}

// ---------------------------------------------------------------------------
// 4) Layer 1: (M x 12) @ (12 x 128) + bias/BN/ReLU, V_WMMA_F32_16X16X4_F32.
//    One 16x16 output tile per wave; 8 waves/block cover 16 rows x 128 cols.
//    Fragment layout (ISA 7.12.2, wave32):
//      A 16x4 : lane l -> row l&15, K = (l>>4)*2 + {0,1} in float2
//      B 4x16 : lane l -> col l&15, K = (l>>4)*2 + {0,1} in float2
//      C/D    : c[r] -> row r + (l>>4)*8, col l&15
// ---------------------------------------------------------------------------
__global__ void __launch_bounds__(256)
gemm1_kernel(const float* __restrict__ A, const float* __restrict__ W,
             const float* __restrict__ b1, const float* __restrict__ g1,
             const float* __restrict__ be1, const float* __restrict__ m1,
             const float* __restrict__ v1, float* __restrict__ Y) {
  const int lane  = threadIdx.x & 31;
  const int wave  = threadIdx.x >> 5;     // 0..7 -> column tile
  const int m0    = blockIdx.x * 16;
  const int n0    = wave * 16;
  const int l15   = lane & 15;
  const int khalf = (lane >> 4) * 2;      // 0 or 2

  v8f c = {};
#pragma unroll
  for (int kc = 0; kc < 3; ++kc) {        // K = 12 = 3 chunks of 4
    const int kb = kc * 4 + khalf;
    v2f a, b;
    a.x = A[(size_t)(m0 + l15) * 12 + kb + 0];
    a.y = A[(size_t)(m0 + l15) * 12 + kb + 1];
    b.x = W[(size_t)(kb + 0) * E_F + n0 + l15];
    b.y = W[(size_t)(kb + 1) * E_F + n0 + l15];
    c = __builtin_amdgcn_wmma_f32_16x16x4_f32(false, a, false, b,
                                              (short)0, c, false, false);
  }

  const int   col   = n0 + l15;
  const float scale = g1[col] * rsqrtf(v1[col] + EPS_BN);
  const float shift = be1[col] + (b1[col] - m1[col]) * scale;
  const int   rbase = m0 + ((lane >> 4) << 3);
#pragma unroll
  for (int r = 0; r < 8; ++r) {
    float y = c[r] * scale + shift;
    Y[(size_t)(rbase + r) * E_F + col] = fmaxf(y, 0.0f);
  }
}

// ---------------------------------------------------------------------------
// 5) Layer 2: (M x 128) @ (128 x 128) + bias/BN/ReLU, f32 WMMA.
//    A-strip (16x128, 8 KB) staged in LDS once per block, reused by 8 waves.
//    K = 32 chunks split into TWO independent accumulator chains so the
//    matrix pipe sees 2-way ILP instead of one 32-deep serial D->C chain.
// ---------------------------------------------------------------------------
__global__ void __launch_bounds__(256)
gemm2_kernel(const float* __restrict__ X, const float* __restrict__ W,
             const float* __restrict__ b2, const float* __restrict__ g2,
             const float* __restrict__ be2, const float* __restrict__ m2,
             const float* __restrict__ v2, float* __restrict__ Out) {
  __shared__ float sA[16 * E_F];          // 8 KB
  const int tid  = threadIdx.x;
  const int lane = tid & 31;
  const int wave = tid >> 5;
  const int m0   = blockIdx.x * 16;

  // Cooperative 128-bit staged load of the 16x128 A-strip.
  const float4* src = (const float4*)(X + (size_t)m0 * E_F);
  float4*       dst = (float4*)sA;
  dst[tid]       = src[tid];
  dst[tid + 256] = src[tid + 256];
  // Hint the next block's strip toward the caches (global_prefetch_b8).
  {
    int nb = blockIdx.x + 1;
    if (nb >= (int)gridDim.x) nb = blockIdx.x;
    __builtin_prefetch(X + (size_t)nb * 16 * E_F, 0, 1);
  }
  __syncthreads();

  const int n0    = wave * 16;
  const int l15   = lane & 15;
  const int khalf = (lane >> 4) * 2;

  v8f c0 = {};
  v8f c1 = {};
#pragma unroll
  for (int kc = 0; kc < 32; kc += 2) {    // K = 128 = 32 chunks of 4
    {
      const int kb = kc * 4 + khalf;
      v2f a, b;
      a.x = sA[l15 * E_F + kb + 0];
      a.y = sA[l15 * E_F + kb + 1];
      b.x = W[(size_t)(kb + 0) * E_F + n0 + l15];
      b.y = W[(size_t)(kb + 1) * E_F + n0 + l15];
      c0 = __builtin_amdgcn_wmma_f32_16x16x4_f32(false, a, false, b,
                                                 (short)0, c0, false, false);
    }
    {
      const int kb = (kc + 1) * 4 + khalf;
      v2f a, b;
      a.x = sA[l15 * E_F + kb + 0];
      a.y = sA[l15 * E_F + kb + 1];
      b.x = W[(size_t)(kb + 0) * E_F + n0 + l15];
      b.y = W[(size_t)(kb + 1) * E_F + n0 + l15];
      c1 = __builtin_amdgcn_wmma_f32_16x16x4_f32(false, a, false, b,
                                                 (short)0, c1, false, false);
    }
  }
  v8f c = c0 + c1;

  const int   col   = n0 + l15;
  const float scale = g2[col] * rsqrtf(v2[col] + EPS_BN);
  const float shift = be2[col] + (b2[col] - m2[col]) * scale;
  const int   rbase = m0 + ((lane >> 4) << 3);
#pragma unroll
  for (int r = 0; r < 8; ++r) {
    float y = c[r] * scale + shift;
    Out[(size_t)(rbase + r) * E_F + col] = fmaxf(y, 0.0f);
  }
}

// ---------------------------------------------------------------------------
extern "C" void kernel_launch(void* const* d_in, const int* in_sizes, int n_in,
                              void* d_out, int out_size, void* d_ws,
                              size_t ws_size, hipStream_t stream) {
  const float* points = (const float*)d_in[0];
  const float* W1     = (const float*)d_in[1];
  const float* b1     = (const float*)d_in[2];
  const float* g1     = (const float*)d_in[3];
  const float* be1    = (const float*)d_in[4];
  const float* m1     = (const float*)d_in[5];
  const float* v1     = (const float*)d_in[6];
  const float* W2     = (const float*)d_in[7];
  const float* b2     = (const float*)d_in[8];
  const float* g2     = (const float*)d_in[9];
  const float* be2    = (const float*)d_in[10];
  const float* m2     = (const float*)d_in[11];
  const float* v2     = (const float*)d_in[12];
  const int*   first  = (const int*)d_in[13];
  float* out = (float*)d_out;

  char* ws = (char*)d_ws;
  size_t off = 0;
  auto carve = [&](size_t bytes) -> char* {
    char* p = ws + off;
    off = (off + bytes + 255) & ~(size_t)255;
    return p;
  };
  float* xyz    = (float*)carve((size_t)N_PTS * 3 * sizeof(float));
  int*   ids    = (int*)  carve((size_t)S_SAMP * sizeof(int));
  int*   groups = (int*)  carve((size_t)S_SAMP * K_NB * sizeof(int));
  float* feats  = (float*)carve((size_t)M_ROWS * 12 * sizeof(float));
  float* x1     = (float*)carve((size_t)M_ROWS * E_F * sizeof(float));

  pack_xyz_kernel<<<(N_PTS + 255) / 256, 256, 0, stream>>>(points, xyz);
  fps_kernel<<<1, 1024, 0, stream>>>(xyz, ids, first);
  topk_kernel<<<S_SAMP, 256, (size_t)N_PTS * sizeof(float), stream>>>(xyz, ids,
                                                                      groups);
  feats_kernel<<<(M_ROWS + 255) / 256, 256, 0, stream>>>(points, ids, groups,
                                                         feats);
  gemm1_kernel<<<M_ROWS / 16, 256, 0, stream>>>(feats, W1, b1, g1, be1, m1, v1,
                                                x1);
  gemm2_kernel<<<M_ROWS / 16, 256, 0, stream>>>(x1, W2, b2, g2, be2, m2, v2,
                                                out);
}